// DeepseekV2AttentionMatAbsorbDecode_36086315221653
// MI455X (gfx1250) — compile-verified
//
#include <hip/hip_runtime.h>
#include <math.h>

#define H     5120
#define NH    128
#define QLR   1536
#define ROPE  64
#define KVLR  512
#define BSZ   8
#define SCALE 0.07216878364870322f      // 192^-0.5
#define LOG_THETA 9.210340371976184f    // ln(10000)
#define KCH   64                        // split-K chunks, output projection
#define KC1   8                         // split-K chunks, gemm1 (5120/8 = 640)

typedef __attribute__((ext_vector_type(2))) float v2f;
typedef __attribute__((ext_vector_type(8))) float v8f;

__device__ __forceinline__ v8f wmma_f32(v2f a, v2f b, v8f c) {
  // D = A(16x4 f32) * B(4x16 f32) + C(16x16 f32)
  return __builtin_amdgcn_wmma_f32_16x16x4_f32(false, a, false, b, (short)0, c,
                                               false, false);
}

// ---------------- 1) hs @ W_DQ : split-K partials -------------------------
// grid (QLR/256, KC1), block 256. P1[kc][b][n]
__global__ void gemm1_kernel(const float* __restrict__ hs,
                             const float* __restrict__ W,
                             float* __restrict__ P1) {
  __shared__ float sh[BSZ * (H / KC1)];              // 8*640 floats = 20 KB
  const int t  = threadIdx.x;
  const int n  = blockIdx.x * 256 + t;
  const int kc = blockIdx.y;
  const int k0 = kc * (H / KC1);
  for (int i = t; i < BSZ * (H / KC1); i += 256) {
    int b = i / (H / KC1), k = i % (H / KC1);
    sh[i] = hs[b * H + k0 + k];
  }
  __syncthreads();
  float acc[BSZ];
#pragma unroll
  for (int b = 0; b < BSZ; ++b) acc[b] = 0.0f;
  for (int k = 0; k < H / KC1; ++k) {
    float w = W[(long)(k0 + k) * QLR + n];
#pragma unroll
    for (int b = 0; b < BSZ; ++b) acc[b] = fmaf(sh[b * (H / KC1) + k], w, acc[b]);
  }
#pragma unroll
  for (int b = 0; b < BSZ; ++b)
    P1[((long)kc * BSZ + b) * QLR + n] = acc[b];
}

// ---------------- 2) reduce partials + RMSNorm -> cQ ----------------------
// grid BSZ, block 256
__global__ void rmsnorm_kernel(const float* __restrict__ P1,
                               const float* __restrict__ lnw,
                               float* __restrict__ cQ) {
  const int b = blockIdx.x, t = threadIdx.x;
  float y[QLR / 256];
  float ssq = 0.0f;
#pragma unroll
  for (int j = 0; j < QLR / 256; ++j) {
    int n = t + j * 256;
    float v = 0.0f;
#pragma unroll
    for (int kc = 0; kc < KC1; ++kc) v += P1[((long)kc * BSZ + b) * QLR + n];
    y[j] = v;
    ssq += v * v;
  }
  __shared__ float red[256];
  red[t] = ssq;
  __syncthreads();
  for (int s = 128; s > 0; s >>= 1) {
    if (t < s) red[t] += red[t + s];
    __syncthreads();
  }
  float inv = rsqrtf(red[0] / (float)QLR + 1e-6f);
#pragma unroll
  for (int j = 0; j < QLR / 256; ++j) {
    int n = t + j * 256;
    cQ[b * QLR + n] = lnw[n] * y[j] * inv;
  }
}

// ---------------- 3) q_pe = cQ @ W_QR, fused RoPE(last pos) ---------------
// grid (NH*ROPE/256), block 256
__global__ void qpe_kernel(const float* __restrict__ cQ,
                           const float* __restrict__ W,
                           float* __restrict__ qp, int kv_len) {
  __shared__ float sh[BSZ * QLR];                    // 48 KB
  __shared__ float ex[BSZ][256];                     // 8 KB pair exchange
  const int t = threadIdx.x;
  for (int i = t; i < BSZ * QLR; i += 256) sh[i] = cQ[i];
  __syncthreads();
  const int n = blockIdx.x * 256 + t;
  float acc[BSZ];
#pragma unroll
  for (int b = 0; b < BSZ; ++b) acc[b] = 0.0f;
  for (int k = 0; k < QLR; ++k) {
    float w = W[(long)k * (NH * ROPE) + n];
#pragma unroll
    for (int b = 0; b < BSZ; ++b) acc[b] = fmaf(sh[b * QLR + k], w, acc[b]);
  }
#pragma unroll
  for (int b = 0; b < BSZ; ++b) ex[b][t] = acc[b];
  __syncthreads();
  const int r = n & (ROPE - 1);
  const int i = r >> 1;
  float inv = __expf(-LOG_THETA * (2.0f * (float)i) / (float)ROPE);
  float ang = (float)(kv_len - 1) * inv;
  float c = cosf(ang), s = sinf(ang);
  const int partner = t ^ 1;
#pragma unroll
  for (int b = 0; b < BSZ; ++b) {
    float self = ex[b][t], other = ex[b][partner];
    float o = (r & 1) ? (other * s + self * c) : (self * c - other * s);
    qp[(long)b * (NH * ROPE) + n] = o;
  }
}

// ---------------- 4) q_nope = cQ @ W_UQ_UK --------------------------------
// grid (NH*KVLR/256), block 256; streams the 402 MB weight once, coalesced
__global__ void qnope_kernel(const float* __restrict__ cQ,
                             const float* __restrict__ W,
                             float* __restrict__ qn) {
  __shared__ float sh[BSZ * QLR];                    // 48 KB
  const int t = threadIdx.x;
  for (int i = t; i < BSZ * QLR; i += 256) sh[i] = cQ[i];
  __syncthreads();
  const long n = (long)blockIdx.x * 256 + t;
  float acc[BSZ];
#pragma unroll
  for (int b = 0; b < BSZ; ++b) acc[b] = 0.0f;
  for (int k = 0; k < QLR; ++k) {
    float w = W[(long)k * (NH * KVLR) + n];
#pragma unroll
    for (int b = 0; b < BSZ; ++b) acc[b] = fmaf(sh[b * QLR + k], w, acc[b]);
  }
#pragma unroll
  for (int b = 0; b < BSZ; ++b)
    qn[(long)b * (NH * KVLR) + n] = acc[b];
}

// ---------------- 5) RoPE on k_pe cache -----------------------------------
__global__ void rope_kpe_kernel(const float* __restrict__ kpe,
                                float* __restrict__ out, int kv_len) {
  long idx = (long)blockIdx.x * 256 + threadIdx.x;   // one rotation pair
  long total = (long)BSZ * kv_len * (ROPE / 2);
  if (idx >= total) return;
  int i = (int)(idx % (ROPE / 2));
  int k = (int)((idx / (ROPE / 2)) % kv_len);
  int b = (int)(idx / ((long)(ROPE / 2) * kv_len));
  float inv = __expf(-LOG_THETA * (2.0f * (float)i) / (float)ROPE);
  float ang = (float)k * inv;
  float c = cosf(ang), s = sinf(ang);
  const float* src = kpe + ((long)b * kv_len + k) * ROPE;
  float* dst = out + ((long)b * kv_len + k) * ROPE;
  float x0 = src[2 * i], x1 = src[2 * i + 1];
  dst[2 * i]     = x0 * c - x1 * s;
  dst[2 * i + 1] = x0 * s + x1 * c;
}

// ---------------- 6) scores = q_pe.k_pe^T + q_nope.ckv^T (WMMA) -----------
// grid (kv_len/128, NH/16, BSZ), block 256 (8 waves, one 16-wide kv tile each)
__global__ void scores_kernel(const float* __restrict__ qn,
                              const float* __restrict__ qp,
                              const float* __restrict__ ckv,
                              const float* __restrict__ kper,
                              float* __restrict__ S, int kv_len) {
  const int lane = threadIdx.x & 31;
  const int wave = threadIdx.x >> 5;
  const int b  = blockIdx.z;
  const int h0 = blockIdx.y * 16;
  const int k0 = blockIdx.x * 128 + wave * 16;
  const int l15 = lane & 15;
  const int kb  = (lane >> 4) * 2;
  const float* qn_row  = qn  + ((long)b * NH + h0 + l15) * KVLR;
  const float* ckv_row = ckv + ((long)b * kv_len + k0 + l15) * KVLR;
  const float* qp_row  = qp  + ((long)b * NH + h0 + l15) * ROPE;
  const float* kpe_row = kper + ((long)b * kv_len + k0 + l15) * ROPE;
  v8f acc = {0.f, 0.f, 0.f, 0.f, 0.f, 0.f, 0.f, 0.f};
#pragma unroll 4
  for (int c = 0; c < KVLR; c += 4) {                // latent part, K=512
    v2f a, bb;
    a.x  = qn_row[c + kb];  a.y  = qn_row[c + kb + 1];
    bb.x = ckv_row[c + kb]; bb.y = ckv_row[c + kb + 1];
    acc = wmma_f32(a, bb, acc);
  }
#pragma unroll 4
  for (int c = 0; c < ROPE; c += 4) {                // rope part, K=64
    v2f a, bb;
    a.x  = qp_row[c + kb];  a.y  = qp_row[c + kb + 1];
    bb.x = kpe_row[c + kb]; bb.y = kpe_row[c + kb + 1];
    acc = wmma_f32(a, bb, acc);
  }
#pragma unroll
  for (int r = 0; r < 8; ++r) {
    int m = h0 + r + 8 * (lane >> 4);
    S[((long)b * NH + m) * kv_len + k0 + l15] = acc[r] * SCALE;
  }
}

// ---------------- 7) softmax over kv (in place) ---------------------------
// grid BSZ*NH, block 256
__global__ void softmax_kernel(float* __restrict__ S, int kv_len) {
  float* x = S + (long)blockIdx.x * kv_len;
  const int t = threadIdx.x;
  __shared__ float red[256];
  float m = -3.4e38f;
  for (int i = t; i < kv_len; i += 256) m = fmaxf(m, x[i]);
  red[t] = m;
  __syncthreads();
  for (int s = 128; s > 0; s >>= 1) {
    if (t < s) red[t] = fmaxf(red[t], red[t + s]);
    __syncthreads();
  }
  const float mx = red[0];
  __syncthreads();
  float sum = 0.0f;
  for (int i = t; i < kv_len; i += 256) {
    float p = __expf(x[i] - mx);
    x[i] = p;
    sum += p;
  }
  red[t] = sum;
  __syncthreads();
  for (int s = 128; s > 0; s >>= 1) {
    if (t < s) red[t] += red[t + s];
    __syncthreads();
  }
  const float invs = 1.0f / red[0];
  for (int i = t; i < kv_len; i += 256) x[i] *= invs;
}

// ---------------- 8) attn = probs @ ckv (WMMA, K=kv_len) ------------------
// grid (KVLR/16, 1, BSZ), block 256 (8 waves = 8 head tiles)
__global__ void attn_kernel(const float* __restrict__ P,
                            const float* __restrict__ ckv,
                            float* __restrict__ A, int kv_len) {
  const int lane = threadIdx.x & 31;
  const int h0   = (threadIdx.x >> 5) * 16;
  const int b    = blockIdx.z;
  const int n0   = blockIdx.x * 16;
  const int l15  = lane & 15;
  const int kb   = (lane >> 4) * 2;
  const float* Prow = P + ((long)b * NH + h0 + l15) * kv_len;
  const float* Cb   = ckv + (long)b * kv_len * KVLR + n0 + l15;
  v8f acc = {0.f, 0.f, 0.f, 0.f, 0.f, 0.f, 0.f, 0.f};
#pragma unroll 4
  for (int c = 0; c < kv_len; c += 4) {
    v2f a, bb;
    a.x  = Prow[c + kb];  a.y = Prow[c + kb + 1];
    bb.x = Cb[(long)(c + kb) * KVLR];
    bb.y = Cb[(long)(c + kb + 1) * KVLR];
    acc = wmma_f32(a, bb, acc);
  }
#pragma unroll
  for (int r = 0; r < 8; ++r) {
    int m = h0 + r + 8 * (lane >> 4);
    A[((long)b * NH + m) * KVLR + n0 + l15] = acc[r];
  }
}

// ---------------- 9) out partials = attn_flat @ W_UV_O (WMMA, split-K) ----
// grid (H/16, KCH), block 32. M=8 rows padded to 16 with zeros.
__global__ void outproj_kernel(const float* __restrict__ A,
                               const float* __restrict__ W,
                               float* __restrict__ P2) {
  const int lane = threadIdx.x;
  const int n0   = blockIdx.x * 16;
  const int kc   = blockIdx.y;
  const int kspan  = (NH * KVLR) / KCH;              // 1024
  const int kstart = kc * kspan;
  const int m   = lane & 15;
  const int kb  = (lane >> 4) * 2;
  const bool valid = (m < BSZ);
  const float* Arow = A + (long)m * (NH * KVLR);
  v8f acc = {0.f, 0.f, 0.f, 0.f, 0.f, 0.f, 0.f, 0.f};
#pragma unroll 4
  for (int c = 0; c < kspan; c += 4) {
    int k = kstart + c + kb;
    v2f a, bb;
    a.x = valid ? Arow[k]     : 0.0f;
    a.y = valid ? Arow[k + 1] : 0.0f;
    bb.x = W[(long)k * H + n0 + (lane & 15)];
    bb.y = W[(long)(k + 1) * H + n0 + (lane & 15)];
    acc = wmma_f32(a, bb, acc);
  }
  if (lane < 16) {                                   // rows 0..7 live here
#pragma unroll
    for (int r = 0; r < BSZ; ++r)
      P2[((long)kc * BSZ + r) * H + n0 + lane] = acc[r];
  }
}

// ---------------- 10) deterministic split-K reduction ---------------------
__global__ void reduce_kernel(const float* __restrict__ P2,
                              float* __restrict__ out) {
  int idx = blockIdx.x * 256 + threadIdx.x;
  if (idx >= BSZ * H) return;
  float s = 0.0f;
#pragma unroll 8
  for (int kc = 0; kc < KCH; ++kc) s += P2[(long)kc * BSZ * H + idx];
  out[idx] = s;
}

extern "C" void kernel_launch(void* const* d_in, const int* in_sizes, int n_in,
                              void* d_out, int out_size, void* d_ws, size_t ws_size,
                              hipStream_t stream) {
  const float* hs   = (const float*)d_in[0];
  const float* ckv  = (const float*)d_in[1];
  const float* kpe  = (const float*)d_in[2];
  const float* wdq  = (const float*)d_in[3];
  const float* lnw  = (const float*)d_in[4];
  const float* wqr  = (const float*)d_in[5];
  const float* wuqk = (const float*)d_in[6];
  const float* wuvo = (const float*)d_in[7];
  float* out = (float*)d_out;

  const int kv_len = in_sizes[1] / (BSZ * KVLR);     // 8192

  // Workspace carve-up (floats): ~66 MB total
  float* ws = (float*)d_ws;
  size_t o = 0;
  float* P1   = ws + o; o += (size_t)KC1 * BSZ * QLR;        // gemm1 partials
  float* cQ   = ws + o; o += (size_t)BSZ * QLR;
  float* qp   = ws + o; o += (size_t)BSZ * NH * ROPE;
  float* qn   = ws + o; o += (size_t)BSZ * NH * KVLR;
  float* kper = ws + o; o += (size_t)BSZ * kv_len * ROPE;    // rotated k_pe
  float* S    = ws + o; o += (size_t)BSZ * NH * kv_len;      // scores/probs
  float* At   = ws + o; o += (size_t)BSZ * NH * KVLR;        // attn latent
  float* P2   = ws + o; o += (size_t)KCH * BSZ * H;          // outproj partials

  gemm1_kernel<<<dim3(QLR / 256, KC1), 256, 0, stream>>>(hs, wdq, P1);
  rmsnorm_kernel<<<BSZ, 256, 0, stream>>>(P1, lnw, cQ);
  qpe_kernel<<<dim3((NH * ROPE) / 256), 256, 0, stream>>>(cQ, wqr, qp, kv_len);
  qnope_kernel<<<dim3((NH * KVLR) / 256), 256, 0, stream>>>(cQ, wuqk, qn);
  {
    long pairs = (long)BSZ * kv_len * (ROPE / 2);
    rope_kpe_kernel<<<dim3((unsigned)((pairs + 255) / 256)), 256, 0, stream>>>(kpe, kper, kv_len);
  }
  scores_kernel<<<dim3(kv_len / 128, NH / 16, BSZ), 256, 0, stream>>>(qn, qp, ckv, kper, S, kv_len);
  softmax_kernel<<<dim3(BSZ * NH), 256, 0, stream>>>(S, kv_len);
  attn_kernel<<<dim3(KVLR / 16, 1, BSZ), 256, 0, stream>>>(S, ckv, At, kv_len);
  outproj_kernel<<<dim3(H / 16, KCH), 32, 0, stream>>>(At, wuvo, P2);
  reduce_kernel<<<dim3((BSZ * H + 255) / 256), 256, 0, stream>>>(P2, out);
}